// DynamicsEmbedding_38749194944562
// MI455X (gfx1250) — compile-verified
//
#include <hip/hip_runtime.h>
#include <hip/hip_bf16.h>
#include <math.h>

// ---------------------------------------------------------------------------
// FNO spectral conv for MI455X (gfx1250, wave32, WMMA), truncated-DFT-as-
// matmul in fp32 via V_WMMA_F32_16X16X4_F32.  B=8, H=W=256, C=64, M1=M2=32.
//
// Round-2 changes (instruction economy; VMEM-issue is the limiter):
//  * 2x2 register tiling per wave  -> each operand fragment feeds 2 WMMAs
//  * all K-major operands pair-interleaved [k/2][M][2] -> one b64 per fragment
//  => loads/WMMA: 4.0 -> 1.0-1.5
// ---------------------------------------------------------------------------

typedef float v2f __attribute__((ext_vector_type(2)));
typedef float v8f __attribute__((ext_vector_type(8)));

__device__ __forceinline__ v8f wmma4(v2f a, v2f b, v8f c) {
  return __builtin_amdgcn_wmma_f32_16x16x4_f32(false, a, false, b, (short)0, c,
                                               false, false);
}
__device__ __forceinline__ v2f ldv2(const float* p) {
  return *(const v2f*)p;  // 8B-aligned by construction
}

// ---- workspace layout (float offsets), ~118.5 MB total ---------------------
constexpr size_t TW1_OFF = 0;                 // [128 kp][64 m][2]     16384
constexpr size_t TW2_OFF = 16384;             // [256 kp][128 m'][2]   65536
constexpr size_t TW4_OFF = 81920;             // [64 kp][512 row][2]   65536
constexpr size_t TW5_OFF = 147456;            // [32 kp][256 w][2]     16384
constexpr size_t S1_OFF  = 262144;            // [8][256 q][2048][2] = 8388608 (also G)
constexpr size_t S2_OFF  = S1_OFF + 8388608;  // [8][128 m'][2048]   = 2097152
constexpr size_t S3_OFF  = S2_OFF + 2097152;  // [8][64 q][2048][2]  = 2097152
constexpr size_t WRT_OFF = S3_OFF + 2097152;  // [2][1024 mk][32 ip][64 o][2]
constexpr size_t WIT_OFF = WRT_OFF + 8388608; //   = 8388608 each

// ---------------------------------------------------------------------------
// Twiddles.  Exact phase: angle = ((k*x) mod 256) * (2*pi/256).
// Stored pair-interleaved: logical T[k][m] -> ws[((k>>1)*M + m)*2 + (k&1)].
// ---------------------------------------------------------------------------
__global__ __launch_bounds__(256) void init_twiddles(float* __restrict__ ws) {
  const int idx = blockIdx.x * blockDim.x + threadIdx.x;  // 0..65535
  const float STEP = 0.024543692606170259f;               // 2*pi/256

  // tw2: forward H-DFT.  K=kk=(part_in,h) (512), M=m'=(part_out,m) (128)
  {
    int kk = idx >> 7, mp = idx & 127;
    int part_in = kk >> 8, h = kk & 255;
    int part_out = mp >> 6, m = mp & 63;
    int kx = (m < 32) ? m : m + 192;
    int t = (kx * h) & 255;
    float c = cosf(t * STEP), s = sinf(t * STEP);
    // Yr = sum Xr*cos + Xi*sin ; Yi = sum -Xr*sin + Xi*cos
    float v = (part_out == 0) ? ((part_in == 0) ? c : s)
                              : ((part_in == 0) ? -s : c);
    ws[TW2_OFF + (size_t)((kk >> 1) * 128 + mp) * 2 + (kk & 1)] = v;
  }
  // tw4: inverse H-DFT /256.  K=k'=(part_in,m) (128), M=row=(part_out,h) (512)
  {
    int k = idx >> 9, row = idx & 511;
    int part_in = k >> 6, m = k & 63;
    int kx = (m < 32) ? m : m + 192;
    int part_out = row >> 8, h = row & 255;
    int t = (kx * h) & 255;
    float c = cosf(t * STEP), s = sinf(t * STEP);
    // Gr = (Zr*cos - Zi*sin)/256 ; Gi = (Zr*sin + Zi*cos)/256
    float v = (part_out == 0) ? ((part_in == 0) ? c : -s)
                              : ((part_in == 0) ? s : c);
    ws[TW4_OFF + (size_t)((k >> 1) * 512 + row) * 2 + (k & 1)] = v * (1.0f / 256.0f);
  }
  if (idx < 16384) {
    // tw1: forward W-rDFT.  K=w (256), M=(part,ky) (64).  Xr=+cos, Xi=-sin
    {
      int w = idx >> 6, mrow = idx & 63;
      int part = mrow >> 5, ky = mrow & 31;
      int t = (ky * w) & 255;
      float v = (part == 0) ? cosf(t * STEP) : -sinf(t * STEP);
      ws[TW1_OFF + (size_t)((w >> 1) * 64 + mrow) * 2 + (w & 1)] = v;
    }
    // tw5: inverse real W-DFT, alpha=(ky==0?1:2)/256 folded. K=(part,ky), M=w
    {
      int k = idx >> 8, w = idx & 255;
      int part = k >> 5, ky = k & 31;
      int t = (ky * w) & 255;
      float alpha = ((ky == 0) ? 1.0f : 2.0f) / 256.0f;
      float v = (part == 0) ? alpha * cosf(t * STEP) : -alpha * sinf(t * STEP);
      ws[TW5_OFF + (size_t)((k >> 1) * 256 + w) * 2 + (k & 1)] = v;
    }
  }
}

// ---------------------------------------------------------------------------
// Weight repack: in w[i,o,mm,ky] (= [io][mk]) -> out[mk][i/2][o][2]
// ---------------------------------------------------------------------------
__global__ __launch_bounds__(1024) void repack_weights(
    const float* __restrict__ in, float* __restrict__ out) {
  __shared__ float tile[32][33];
  int x = blockIdx.x * 32 + threadIdx.x;  // mk
  int y = blockIdx.y * 32 + threadIdx.y;  // io
  tile[threadIdx.y][threadIdx.x] = in[(size_t)y * 1024 + x];
  __syncthreads();
  int ox = blockIdx.y * 32 + threadIdx.x;  // io = i*64 + o
  int oy = blockIdx.x * 32 + threadIdx.y;  // mk
  int off = ((ox >> 7) << 7) + ((ox & 63) << 1) + ((ox >> 6) & 1);
  out[(size_t)oy * 4096 + off] = tile[threadIdx.x][threadIdx.y];
}

// ---------------------------------------------------------------------------
// Stage 1: per (b,h)  D[64x64] = TW1^T(64x256) * u[b,h](256x64)
// 4 waves/block, each wave a 32x32 super-tile (2x2 WMMA tiles).
// out s1i[b][kk/2][ky*64+c][kk&1], kk=(part,h)
// ---------------------------------------------------------------------------
__global__ __launch_bounds__(128) void s1_dft_w(const float* __restrict__ u,
                                                float* __restrict__ ws) {
  const float* tw = ws + TW1_OFF;
  float* s1 = ws + S1_OFF;
  const int wave = threadIdx.x >> 5, lane = threadIdx.x & 31;
  const int half = lane >> 4, l16 = lane & 15;
  const int bh = blockIdx.x;
  const int b = bh >> 8, h = bh & 255;
  const float* ubh = u + (size_t)bh * (256 * 64);
  const int mT2 = wave >> 1, nT2 = wave & 1;
  const int mrow0 = mT2 * 32 + l16, mrow1 = mrow0 + 16;
  const int col0 = nT2 * 32 + l16, col1 = col0 + 16;
  v8f a00 = {}, a01 = {}, a10 = {}, a11 = {};
#pragma unroll 4
  for (int ks = 0; ks < 64; ++ks) {
    const int k = ks * 4 + half * 2, kp = ks * 2 + half;
    v2f A0 = ldv2(tw + (size_t)(kp * 64 + mrow0) * 2);
    v2f A1 = ldv2(tw + (size_t)(kp * 64 + mrow1) * 2);
    v2f B0, B1;
    B0.x = ubh[(size_t)k * 64 + col0];
    B0.y = ubh[(size_t)(k + 1) * 64 + col0];
    B1.x = ubh[(size_t)k * 64 + col1];
    B1.y = ubh[(size_t)(k + 1) * 64 + col1];
    a00 = wmma4(A0, B0, a00); a01 = wmma4(A0, B1, a01);
    a10 = wmma4(A1, B0, a10); a11 = wmma4(A1, B1, a11);
  }
  const int q0 = b * 262144;  // b*512*2048/4? no: float offset per b = 1048576
  (void)q0;
  float* s1b = s1 + (size_t)b * 1048576;
#pragma unroll
  for (int r = 0; r < 8; ++r) {
    const int m0 = mT2 * 32 + half * 8 + r;       // (part,ky) of sub-tile 0
    const int m1 = m0 + 16;                       // sub-tile 1
    const int p0 = m0 >> 5, ky0 = m0 & 31;
    const int p1 = m1 >> 5, ky1 = m1 & 31;
    const size_t q_0 = (size_t)(p0 * 128 + (h >> 1));
    const size_t q_1 = (size_t)(p1 * 128 + (h >> 1));
    const int j = h & 1;
    s1b[(q_0 * 2048 + ky0 * 64 + col0) * 2 + j] = a00[r];
    s1b[(q_0 * 2048 + ky0 * 64 + col1) * 2 + j] = a01[r];
    s1b[(q_1 * 2048 + ky1 * 64 + col0) * 2 + j] = a10[r];
    s1b[(q_1 * 2048 + ky1 * 64 + col1) * 2 + j] = a11[r];
  }
}

// ---------------------------------------------------------------------------
// Stage 2: per b  D[128x2048] = TW2(128x512) * s1[b](512x2048)
// out s2[b][m'][ky*64+c]  (not interleaved: S3 reads along contiguous cin)
// ---------------------------------------------------------------------------
__global__ __launch_bounds__(256) void s2_dft_h(float* __restrict__ ws) {
  const float* tw = ws + TW2_OFF;
  const float* s1 = ws + S1_OFF;
  float* s2 = ws + S2_OFF;
  const int wave = threadIdx.x >> 5, lane = threadIdx.x & 31;
  const int half = lane >> 4, l16 = lane & 15;
  const int sid = blockIdx.x * 8 + wave;  // 0..2047
  const int b = sid >> 8;
  const int rem = sid & 255;
  const int mT2 = rem >> 6, nT2 = rem & 63;
  const int mrow0 = mT2 * 32 + l16, mrow1 = mrow0 + 16;
  const int col0 = nT2 * 32 + l16, col1 = col0 + 16;
  const float* s1b = s1 + (size_t)b * 1048576;
  v8f a00 = {}, a01 = {}, a10 = {}, a11 = {};
#pragma unroll 4
  for (int ks = 0; ks < 128; ++ks) {
    const int kp = ks * 2 + half;
    v2f A0 = ldv2(tw + (size_t)(kp * 128 + mrow0) * 2);
    v2f A1 = ldv2(tw + (size_t)(kp * 128 + mrow1) * 2);
    v2f B0 = ldv2(s1b + ((size_t)kp * 2048 + col0) * 2);
    v2f B1 = ldv2(s1b + ((size_t)kp * 2048 + col1) * 2);
    a00 = wmma4(A0, B0, a00); a01 = wmma4(A0, B1, a01);
    a10 = wmma4(A1, B0, a10); a11 = wmma4(A1, B1, a11);
  }
#pragma unroll
  for (int r = 0; r < 8; ++r) {
    const int m0 = mT2 * 32 + half * 8 + r, m1 = m0 + 16;
    s2[((size_t)b * 128 + m0) * 2048 + col0] = a00[r];
    s2[((size_t)b * 128 + m0) * 2048 + col1] = a01[r];
    s2[((size_t)b * 128 + m1) * 2048 + col0] = a10[r];
    s2[((size_t)b * 128 + m1) * 2048 + col1] = a11[r];
  }
}

// ---------------------------------------------------------------------------
// Stage 3: per (m,ky): complex channel mix.  M=16 (batch 0..7 valid), K=cin,
// N = 2x16 couts per wave.  Zr = Yr*Wr - Yi*Wi ; Zi = Yr*Wi + Yi*Wr.
// out s3i[b][m'/2][ky*64+o][m'&1], m'=(part,m)
// ---------------------------------------------------------------------------
__global__ __launch_bounds__(256) void s3_spectral(float* __restrict__ ws) {
  const float* s2 = ws + S2_OFF;
  float* s3 = ws + S3_OFF;
  const float* wrT = ws + WRT_OFF;
  const float* wiT = ws + WIT_OFF;
  const int wave = threadIdx.x >> 5, lane = threadIdx.x & 31;
  const int half = lane >> 4, l16 = lane & 15;
  const int sid = blockIdx.x * 8 + wave;  // 0..4095
  const int m = sid >> 6;
  const int rem = sid & 63;
  const int ky = rem >> 1, nT2 = rem & 1;
  const int src = m >> 5, mm = m & 31;
  const size_t wrow = (size_t)(src * 1024 + mm * 32 + ky) * 4096;
  const float* wr = wrT + wrow;
  const float* wi = wiT + wrow;
  const int o0 = nT2 * 32 + l16, o1 = o0 + 16;
  const bool avalid = (l16 < 8);
  const float* yr = s2 + ((size_t)(l16 * 128) + m) * 2048 + ky * 64;
  const float* yi = s2 + ((size_t)(l16 * 128) + 64 + m) * 2048 + ky * 64;
  v8f r0 = {}, i0 = {}, r1 = {}, i1 = {};
#pragma unroll 2
  for (int ks = 0; ks < 16; ++ks) {
    const int k = ks * 4 + half * 2, kp = ks * 2 + half;
    v2f ar = {0.0f, 0.0f}, ai = {0.0f, 0.0f};
    if (avalid) { ar = ldv2(yr + k); ai = ldv2(yi + k); }
    v2f br0 = ldv2(wr + (size_t)kp * 128 + o0 * 2);
    v2f bi0 = ldv2(wi + (size_t)kp * 128 + o0 * 2);
    v2f br1 = ldv2(wr + (size_t)kp * 128 + o1 * 2);
    v2f bi1 = ldv2(wi + (size_t)kp * 128 + o1 * 2);
    v2f nbi0 = {-bi0.x, -bi0.y}, nbi1 = {-bi1.x, -bi1.y};
    r0 = wmma4(ar, br0, r0); r0 = wmma4(ai, nbi0, r0);
    i0 = wmma4(ar, bi0, i0); i0 = wmma4(ai, br0, i0);
    r1 = wmma4(ar, br1, r1); r1 = wmma4(ai, nbi1, r1);
    i1 = wmma4(ar, bi1, i1); i1 = wmma4(ai, br1, i1);
  }
  if (half == 0) {  // batches 0..7 live in lanes 0-15, VGPR r = batch r
    const int jm = m & 1;
    const size_t qr = (size_t)(m >> 1) * 2048;          // Zr: m' = m
    const size_t qi = (size_t)(32 + (m >> 1)) * 2048;   // Zi: m' = 64+m
#pragma unroll
    for (int r = 0; r < 8; ++r) {
      float* s3b = s3 + (size_t)r * 262144;
      s3b[(qr + ky * 64 + o0) * 2 + jm] = r0[r];
      s3b[(qr + ky * 64 + o1) * 2 + jm] = r1[r];
      s3b[(qi + ky * 64 + o0) * 2 + jm] = i0[r];
      s3b[(qi + ky * 64 + o1) * 2 + jm] = i1[r];
    }
  }
}

// ---------------------------------------------------------------------------
// Stage 4: per b  D[512x2048] = TW4(512x128) * s3[b](128x2048)
// out G[b][(part,h)][ky/2][c][ky&1]  (reuses S1 buffer; ky-pair interleave)
// ---------------------------------------------------------------------------
__global__ __launch_bounds__(256) void s4_idft_h(float* __restrict__ ws) {
  const float* tw = ws + TW4_OFF;
  const float* z = ws + S3_OFF;
  float* g = ws + S1_OFF;
  const int wave = threadIdx.x >> 5, lane = threadIdx.x & 31;
  const int half = lane >> 4, l16 = lane & 15;
  const int sid = blockIdx.x * 8 + wave;  // 0..8191
  const int b = sid >> 10;
  const int rem = sid & 1023;
  const int mT2 = rem >> 6, nT2 = rem & 63;
  const int mrow0 = mT2 * 32 + l16, mrow1 = mrow0 + 16;
  const int col0 = nT2 * 32 + l16, col1 = col0 + 16;
  const float* zb = z + (size_t)b * 262144;
  float* gb = g + (size_t)b * 1048576;
  v8f a00 = {}, a01 = {}, a10 = {}, a11 = {};
#pragma unroll 4
  for (int ks = 0; ks < 32; ++ks) {
    const int kp = ks * 2 + half;
    v2f A0 = ldv2(tw + (size_t)(kp * 512 + mrow0) * 2);
    v2f A1 = ldv2(tw + (size_t)(kp * 512 + mrow1) * 2);
    v2f B0 = ldv2(zb + ((size_t)kp * 2048 + col0) * 2);
    v2f B1 = ldv2(zb + ((size_t)kp * 2048 + col1) * 2);
    a00 = wmma4(A0, B0, a00); a01 = wmma4(A0, B1, a01);
    a10 = wmma4(A1, B0, a10); a11 = wmma4(A1, B1, a11);
  }
  const int ky0 = col0 >> 6, c0 = col0 & 63;
  const int ky1 = col1 >> 6, c1 = col1 & 63;
  const size_t n0 = (size_t)(ky0 >> 1) * 64 + c0;
  const size_t n1 = (size_t)(ky1 >> 1) * 64 + c1;
  const int j0 = ky0 & 1, j1 = ky1 & 1;
#pragma unroll
  for (int r = 0; r < 8; ++r) {
    const int row0 = mT2 * 32 + half * 8 + r, row1 = row0 + 16;
    gb[((size_t)row0 * 1024 + n0) * 2 + j0] = a00[r];
    gb[((size_t)row0 * 1024 + n1) * 2 + j1] = a01[r];
    gb[((size_t)row1 * 1024 + n0) * 2 + j0] = a10[r];
    gb[((size_t)row1 * 1024 + n1) * 2 + j1] = a11[r];
  }
}

// ---------------------------------------------------------------------------
// Stage 5: per (b,h)  out[w,c] = TW5(256x64) * G[b,:,h,:,:](64x64)
// ---------------------------------------------------------------------------
__global__ __launch_bounds__(256) void s5_idft_w(float* __restrict__ ws,
                                                 float* __restrict__ out) {
  const float* tw = ws + TW5_OFF;
  const float* g = ws + S1_OFF;
  const int wave = threadIdx.x >> 5, lane = threadIdx.x & 31;
  const int half = lane >> 4, l16 = lane & 15;
  const int sid = blockIdx.x * 8 + wave;  // 0..32767
  const int bh = sid >> 4;
  const int rem = sid & 15;
  const int mT2 = rem >> 1, nT2 = rem & 1;
  const int b = bh >> 8, h = bh & 255;
  const int wrow0 = mT2 * 32 + l16, wrow1 = wrow0 + 16;
  const int c0 = nT2 * 32 + l16, c1 = c0 + 16;
  const float* gb = g + (size_t)b * 1048576;
  v8f a00 = {}, a01 = {}, a10 = {}, a11 = {};
#pragma unroll 4
  for (int ks = 0; ks < 16; ++ks) {
    const int k = ks * 4 + half * 2, kp = ks * 2 + half;
    const int part = k >> 5, ky = k & 31;  // k even -> (ky,ky+1) same part
    v2f A0 = ldv2(tw + (size_t)(kp * 256 + wrow0) * 2);
    v2f A1 = ldv2(tw + (size_t)(kp * 256 + wrow1) * 2);
    const size_t rb = (size_t)(part * 256 + h) * 1024 + (size_t)(ky >> 1) * 64;
    v2f B0 = ldv2(gb + (rb + c0) * 2);
    v2f B1 = ldv2(gb + (rb + c1) * 2);
    a00 = wmma4(A0, B0, a00); a01 = wmma4(A0, B1, a01);
    a10 = wmma4(A1, B0, a10); a11 = wmma4(A1, B1, a11);
  }
#pragma unroll
  for (int r = 0; r < 8; ++r) {
    const int w0 = mT2 * 32 + half * 8 + r, w1 = w0 + 16;
    out[((size_t)bh * 256 + w0) * 64 + c0] = a00[r];
    out[((size_t)bh * 256 + w0) * 64 + c1] = a01[r];
    out[((size_t)bh * 256 + w1) * 64 + c0] = a10[r];
    out[((size_t)bh * 256 + w1) * 64 + c1] = a11[r];
  }
}

// ---------------------------------------------------------------------------
extern "C" void kernel_launch(void* const* d_in, const int* in_sizes, int n_in,
                              void* d_out, int out_size, void* d_ws,
                              size_t ws_size, hipStream_t stream) {
  (void)in_sizes; (void)n_in; (void)out_size; (void)ws_size;
  const float* u   = (const float*)d_in[0];
  const float* wpr = (const float*)d_in[1];
  const float* wpi = (const float*)d_in[2];
  const float* wnr = (const float*)d_in[3];
  const float* wni = (const float*)d_in[4];
  float* out = (float*)d_out;
  float* ws = (float*)d_ws;

  init_twiddles<<<256, 256, 0, stream>>>(ws);

  dim3 tb(32, 32);
  dim3 tg(1024 / 32, 4096 / 32);
  repack_weights<<<tg, tb, 0, stream>>>(wpr, ws + WRT_OFF);
  repack_weights<<<tg, tb, 0, stream>>>(wnr, ws + WRT_OFF + 4194304);
  repack_weights<<<tg, tb, 0, stream>>>(wpi, ws + WIT_OFF);
  repack_weights<<<tg, tb, 0, stream>>>(wni, ws + WIT_OFF + 4194304);

  s1_dft_w<<<2048, 128, 0, stream>>>(u, ws);
  s2_dft_h<<<256, 256, 0, stream>>>(ws);
  s3_spectral<<<512, 256, 0, stream>>>(ws);
  s4_idft_h<<<1024, 256, 0, stream>>>(ws);
  s5_idft_w<<<4096, 256, 0, stream>>>(ws, out);
}